// SynapticMemory_79164837200282
// MI455X (gfx1250) — compile-verified
//
#include <hip/hip_runtime.h>
#include <hip/hip_bf16.h>
#include <math.h>
#include <stdint.h>

// ---------------------------------------------------------------------------
// SynapticMemory (linear-attention fast-weight scan) for MI455X / gfx1250.
//  - All GEMMs via V_WMMA_F32_16X16X4_F32 (fp32-exact matrix path).
//  - Projection GEMMs: 2x4 register-blocked strips (<1 load per wmma).
//  - Scan: async double-buffered LDS staging (GLOBAL_LOAD_ASYNC_TO_LDS_B128 +
//    S_WAIT_ASYNCCNT), decay folded into K so copies are raw bytes,
//    16-B-aligned padded LDS strides (conflict-free WMMA fragment reads).
// ---------------------------------------------------------------------------

typedef __attribute__((ext_vector_type(2))) float v2f;
typedef __attribute__((ext_vector_type(8))) float v8f;

#define D_MODEL 512
#define MEM_DIM 64
#define BATCH   2
#define SEQ     2048
#define ROWS    (BATCH * SEQ)   // 4096 flattened rows
#define CHUNK   32              // scan chunk length T
#define DBLK    32              // D columns per scan workgroup (32 walkers)
#define NCHUNK  (SEQ / CHUNK)   // 64
#define LDP     68              // stride for 64-wide LDS tiles (16B rows)
#define LDV     36              // stride for 32-wide LDS tiles (16B rows)
#define LDS_SM  33              // stride for the 32-wide S tile

// ---- 16x16 fp32 WMMA tile helpers (ISA 7.12.2 layouts) --------------------
// A fragment: lane l(0..15) holds row M=l; VGPR0/1 = K {0,1} (lanes 0-15)
//             or K {2,3} (lanes 16-31).
// B fragment: lane l(0..15) holds col N=l; VGPR0/1 = K {0,1}/{2,3} by half.
// C/D       : lane = N, VGPR j = row M=j (lanes 0-15) / M=j+8 (lanes 16-31).

__device__ inline v8f wmma_AB(const float* A, int lda,
                              const float* B, int ldb, int K, v8f acc) {
  const int lane = threadIdx.x & 31;
  const int l = lane & 15;
  const int h = (lane >> 4) << 1;     // 0 or 2
  for (int k = 0; k < K; k += 4) {
    v2f a, b;
    a.x = A[l * lda + k + h];
    a.y = A[l * lda + k + h + 1];
    b.x = B[(k + h) * ldb + l];
    b.y = B[(k + h + 1) * ldb + l];
    acc = __builtin_amdgcn_wmma_f32_16x16x4_f32(false, a, false, b,
                                                (short)0, acc, false, false);
  }
  return acc;
}

__device__ inline v8f wmma_ABt(const float* A, int lda,
                               const float* Bt, int ldb, int K, v8f acc) {
  const int lane = threadIdx.x & 31;
  const int l = lane & 15;
  const int h = (lane >> 4) << 1;
  for (int k = 0; k < K; k += 4) {
    v2f a, b;
    a.x = A[l * lda + k + h];
    a.y = A[l * lda + k + h + 1];
    b.x = Bt[l * ldb + k + h];
    b.y = Bt[l * ldb + k + h + 1];
    acc = __builtin_amdgcn_wmma_f32_16x16x4_f32(false, a, false, b,
                                                (short)0, acc, false, false);
  }
  return acc;
}

__device__ inline v8f wmma_AtB(const float* At, int lda,
                               const float* B, int ldb, int K, v8f acc) {
  const int lane = threadIdx.x & 31;
  const int l = lane & 15;
  const int h = (lane >> 4) << 1;
  for (int k = 0; k < K; k += 4) {
    v2f a, b;
    a.x = At[(k + h) * lda + l];
    a.y = At[(k + h + 1) * lda + l];
    b.x = B[(k + h) * ldb + l];
    b.y = B[(k + h + 1) * ldb + l];
    acc = __builtin_amdgcn_wmma_f32_16x16x4_f32(false, a, false, b,
                                                (short)0, acc, false, false);
  }
  return acc;
}

// 32x64 output strip: 2 row tiles x 4 col tiles, shared operand fragments.
__device__ inline void wmma_block_2x4(const float* A, int lda,
                                      const float* Bt, int ldb,
                                      int K, v8f acc[2][4]) {
  const int lane = threadIdx.x & 31;
  const int l = lane & 15;
  const int h = (lane >> 4) << 1;
  for (int k = 0; k < K; k += 4) {
    v2f a[2], b[4];
#pragma unroll
    for (int i = 0; i < 2; ++i) {
      a[i].x = A[(i * 16 + l) * lda + k + h];
      a[i].y = A[(i * 16 + l) * lda + k + h + 1];
    }
#pragma unroll
    for (int j = 0; j < 4; ++j) {
      b[j].x = Bt[(j * 16 + l) * ldb + k + h];
      b[j].y = Bt[(j * 16 + l) * ldb + k + h + 1];
    }
#pragma unroll
    for (int i = 0; i < 2; ++i)
#pragma unroll
      for (int j = 0; j < 4; ++j)
        acc[i][j] = __builtin_amdgcn_wmma_f32_16x16x4_f32(
            false, a[i], false, b[j], (short)0, acc[i][j], false, false);
  }
}

__device__ inline void store_tile(float* D, int ldd, v8f c) {
  const int lane = threadIdx.x & 31;
  const int n = lane & 15;
  const int mo = (lane >> 4) << 3;
#pragma unroll
  for (int j = 0; j < 8; ++j) D[(mo + j) * ldd + n] = c[j];
}

__device__ inline v8f load_tile(const float* S, int lds) {
  const int lane = threadIdx.x & 31;
  const int n = lane & 15;
  const int mo = (lane >> 4) << 3;
  v8f c;
#pragma unroll
  for (int j = 0; j < 8; ++j) c[j] = S[(mo + j) * lds + n];
  return c;
}

__device__ inline float phi_elu1(float t) {        // elu(t)+1
  return t > 0.f ? t + 1.f : __expf(t);
}

__device__ inline unsigned lds_off(const void* p) {   // LDS byte offset
  return (unsigned)(uintptr_t)p;
}

// Async-copy a CHUNK x W float tile (global row stride gs) into LDS at byte
// base ldsBase with row stride LSTRIDE floats.  One b128 per lane per issue;
// tracked on ASYNCcnt.
template<int W, int LSTRIDE>
__device__ inline void issue_async_tile(const float* g, int gs, unsigned ldsBase) {
  const int tid = threadIdx.x;
#pragma unroll
  for (int r = 0; r < (CHUNK * W / 4) / 128; ++r) {
    const int t = tid + 128 * r;
    const int s = (t * 4) / W;
    const int c = (t * 4) % W;
    const unsigned dst = ldsBase + (unsigned)((s * LSTRIDE + c) * 4);
    const float* src = g + (size_t)s * gs + c;
    asm volatile("global_load_async_to_lds_b128 %0, %1, off"
                 :: "v"(dst), "v"(src) : "memory");
  }
}

// ---------------------------------------------------------------------------
// K1: QKV projection.  Col groups of 64: cg0=Q (phi*scale), cg1=K (phi),
// cg2..9=V.  Grid (10, 32), 4 waves/block, one 32x64 strip per wave.
// ---------------------------------------------------------------------------
__global__ void __launch_bounds__(128)
syn_qkv_kernel(const float* __restrict__ x,
               const float* __restrict__ Wq, const float* __restrict__ Wk,
               const float* __restrict__ Wv,
               float* __restrict__ Q, float* __restrict__ K,
               float* __restrict__ V) {
  const int wave = threadIdx.x >> 5;
  const int cg = blockIdx.x;              // 0..9 column groups of 64
  const int rg = blockIdx.y * 4 + wave;   // 0..127 row groups of 32
  const float* A = x + (size_t)rg * 32 * D_MODEL;

  const float* Wt;
  float* out;
  int wn0, ldo, mode;
  if (cg == 0)      { Wt = Wq; wn0 = 0;             out = Q; ldo = MEM_DIM; mode = 0; }
  else if (cg == 1) { Wt = Wk; wn0 = 0;             out = K; ldo = MEM_DIM; mode = 1; }
  else              { Wt = Wv; wn0 = (cg - 2) * 64; out = V; ldo = D_MODEL; mode = 2; }

  v8f acc[2][4];
#pragma unroll
  for (int i = 0; i < 2; ++i)
#pragma unroll
    for (int j = 0; j < 4; ++j) acc[i][j] = (v8f){};

  wmma_block_2x4(A, D_MODEL, Wt + (size_t)wn0 * D_MODEL, D_MODEL, D_MODEL, acc);

  const int lane = threadIdx.x & 31;
  const int n = lane & 15;
  const int mo = (lane >> 4) << 3;
  const float scale = 0.125f;  // MEM_DIM^-0.5
#pragma unroll
  for (int i = 0; i < 2; ++i)
#pragma unroll
    for (int jt = 0; jt < 4; ++jt)
#pragma unroll
      for (int j = 0; j < 8; ++j) {
        float v = acc[i][jt][j];
        if (mode == 0)      v = phi_elu1(v) * scale;
        else if (mode == 1) v = phi_elu1(v);
        out[(size_t)(rg * 32 + i * 16 + mo + j) * ldo + wn0 + jt * 16 + n] = v;
      }
}

// ---------------------------------------------------------------------------
// K2: chunked fast-weight scan.  Grid (16 d-blocks, 2 batches), 128 threads.
// Decay folded into K:  Kw = diag(en) K.  Per chunk (T=32):
//   Sm = mask(Q @ Kw^T)                (carries en)
//   num = diag(ep) * ( Sm @ V + Q @ C )
//   den = ep * (rowsum(Sm) + Q.z);  C += Kw^T @ V;  z += colsum(Kw)
// Q/K/V tiles double-buffered via async LDS copies (prefetch next chunk
// while computing the current one).
// ---------------------------------------------------------------------------
__global__ void __launch_bounds__(128)
syn_scan_kernel(const float* __restrict__ Q, const float* __restrict__ K,
                const float* __restrict__ V,
                const float* __restrict__ decay_logit,
                float* __restrict__ Num, float* __restrict__ Den) {
  __shared__ __align__(16) float sQ[2][CHUNK * LDP];    // 2 x 8704 B
  __shared__ __align__(16) float sKw[2][CHUNK * LDP];   // 2 x 8704 B
  __shared__ __align__(16) float sV[2][CHUNK * LDV];    // 2 x 4608 B
  __shared__ float sSm[CHUNK * LDS_SM];                 // 4224 B
  __shared__ float sC[MEM_DIM * LDV];                   // 9216 B
  __shared__ float sEp[CHUNK];
  __shared__ float sZ[MEM_DIM];

  const int db   = blockIdx.x;              // 0..15 (column block of 32)
  const int b    = blockIdx.y;              // batch
  const int tid  = threadIdx.x;
  const int wave = tid >> 5;

  const float logit = decay_logit[0];
  const float decay = 1.f / (1.f + __expf(-logit));
  const float ld    = __logf(fmaxf(decay, 1e-6f));

  auto issue_chunk = [&](int ch, int buf) {
    const int g0 = ch * CHUNK;
    issue_async_tile<64, LDP>(Q + (size_t)(b * SEQ + g0) * MEM_DIM, MEM_DIM,
                              lds_off(&sQ[buf][0]));
    issue_async_tile<64, LDP>(K + (size_t)(b * SEQ + g0) * MEM_DIM, MEM_DIM,
                              lds_off(&sKw[buf][0]));
    issue_async_tile<32, LDV>(V + (size_t)(b * SEQ + g0) * D_MODEL + db * DBLK,
                              D_MODEL, lds_off(&sV[buf][0]));
  };

  issue_chunk(0, 0);                        // prefetch chunk 0 (10/wave)

  for (int i = tid; i < MEM_DIM * LDV; i += 128) sC[i] = 0.f;
  if (tid < MEM_DIM) sZ[tid] = 0.f;

  for (int ch = 0; ch < NCHUNK; ++ch) {
    const int cur = ch & 1;
    const int g0 = ch * CHUNK;

    if (ch + 1 < NCHUNK) {
      issue_chunk(ch + 1, cur ^ 1);         // prefetch next chunk
      asm volatile("s_wait_asynccnt 0xa" ::: "memory");   // chunk ch landed
    } else {
      asm volatile("s_wait_asynccnt 0x0" ::: "memory");
    }
    __syncthreads();

    // fold decay into K in place; build ep table
    if (tid < CHUNK) {
      float lc = fminf(fmaxf(ld * (float)(g0 + tid), -20.f), 20.f);
      sEp[tid] = __expf(lc);
    }
    for (int i = tid; i < CHUNK * MEM_DIM; i += 128) {
      int s = i >> 6, m = i & 63;
      float lc = fminf(fmaxf(ld * (float)(g0 + s), -20.f), 20.f);
      sKw[cur][s * LDP + m] *= __expf(-lc);
    }
    __syncthreads();

    // ---- Sm = mask( Q @ Kw^T )  : 2x2 tiles, one per wave -----------------
    {
      const int tr = wave >> 1, tc = wave & 1;
      v8f s = {};
      s = wmma_ABt(sQ[cur] + tr * 16 * LDP, LDP,
                   sKw[cur] + tc * 16 * LDP, LDP, MEM_DIM, s);
      const int lane = tid & 31;
      const int n = lane & 15, mo = (lane >> 4) << 3;
#pragma unroll
      for (int j = 0; j < 8; ++j) {
        int trow = tr * 16 + mo + j, scol = tc * 16 + n;
        sSm[trow * LDS_SM + scol] = (scol <= trow) ? s[j] : 0.f;
      }
    }
    __syncthreads();

    // ---- numerator: ep * ( Sm @ V  +  Q @ C_prev ) : 2x2 tiles ------------
    {
      const int tr = wave >> 1, tc = wave & 1;
      v8f acc = {};
      acc = wmma_AB(sSm + tr * 16 * LDS_SM, LDS_SM, sV[cur] + tc * 16, LDV,
                    CHUNK, acc);
      acc = wmma_AB(sQ[cur] + tr * 16 * LDP, LDP, sC + tc * 16, LDV,
                    MEM_DIM, acc);
      const int lane = tid & 31;
      const int n = lane & 15, mo = (lane >> 4) << 3;
#pragma unroll
      for (int j = 0; j < 8; ++j) {
        int t = tr * 16 + mo + j;
        Num[(size_t)(b * SEQ + g0 + t) * D_MODEL + db * DBLK + tc * 16 + n] =
            acc[j] * sEp[t];
      }
    }

    // ---- denominator (one d-block does it) --------------------------------
    if (db == 0 && tid < CHUNK) {
      const int t = tid;
      float rs = 0.f;
      for (int s = 0; s < CHUNK; ++s) rs += sSm[t * LDS_SM + s];
      float qz = 0.f;
      for (int m = 0; m < MEM_DIM; ++m) qz += sQ[cur][t * LDP + m] * sZ[m];
      Den[b * SEQ + g0 + t] = sEp[t] * (rs + qz) + 1e-6f;
    }
    __syncthreads();

    // ---- state update:  C += Kw^T @ V ;  z += colsum(Kw) ------------------
    for (int idx = wave; idx < 8; idx += 4) {    // 4 row x 2 col tiles
      const int tr = idx >> 1, tc = idx & 1;
      float* ctile = sC + tr * 16 * LDV + tc * 16;
      v8f acc = load_tile(ctile, LDV);
      acc = wmma_AtB(sKw[cur] + tr * 16, LDP, sV[cur] + tc * 16, LDV,
                     CHUNK, acc);
      store_tile(ctile, LDV, acc);
    }
    if (tid < MEM_DIM) {
      float a = sZ[tid];
      for (int s = 0; s < CHUNK; ++s) a += sKw[cur][s * LDP + tid];
      sZ[tid] = a;
    }
    __syncthreads();
  }
}

// ---------------------------------------------------------------------------
// K3: u = num/den ; rms-norm ; * norm_w   (one block of 128 per row, wave32)
// ---------------------------------------------------------------------------
__global__ void __launch_bounds__(128)
syn_norm_kernel(const float* __restrict__ Num, const float* __restrict__ Den,
                const float* __restrict__ norm_w, float* __restrict__ Un) {
  __shared__ float red[4];
  const int row = blockIdx.x;
  const int tid = threadIdx.x;
  const float* nr = Num + (size_t)row * D_MODEL;
  const float den = Den[row];
  float u[4];
  float ss = 0.f;
#pragma unroll
  for (int i = 0; i < 4; ++i) {
    float v = nr[tid + i * 128] / den;
    u[i] = v;
    ss += v * v;
  }
#pragma unroll
  for (int off = 16; off > 0; off >>= 1) ss += __shfl_down(ss, off, 32);
  if ((tid & 31) == 0) red[tid >> 5] = ss;
  __syncthreads();
  const float tot = red[0] + red[1] + red[2] + red[3];
  const float inv = rsqrtf(tot * (1.f / D_MODEL) + 1e-6f);
#pragma unroll
  for (int i = 0; i < 4; ++i) {
    int c = tid + i * 128;
    Un[(size_t)row * D_MODEL + c] = u[i] * inv * norm_w[c];
  }
}

// ---------------------------------------------------------------------------
// K4: out = Un @ Wo^T   (grid (8,32), 4 waves, one 32x64 strip per wave)
// ---------------------------------------------------------------------------
__global__ void __launch_bounds__(128)
syn_oproj_kernel(const float* __restrict__ Un, const float* __restrict__ Wo,
                 float* __restrict__ out) {
  const int wave = threadIdx.x >> 5;
  const int cg = blockIdx.x;              // 0..7 column groups of 64
  const int rg = blockIdx.y * 4 + wave;   // 0..127 row groups of 32
  const int wn0 = cg * 64;

  v8f acc[2][4];
#pragma unroll
  for (int i = 0; i < 2; ++i)
#pragma unroll
    for (int j = 0; j < 4; ++j) acc[i][j] = (v8f){};

  wmma_block_2x4(Un + (size_t)rg * 32 * D_MODEL, D_MODEL,
                 Wo + (size_t)wn0 * D_MODEL, D_MODEL, D_MODEL, acc);

  const int lane = threadIdx.x & 31;
  const int n = lane & 15;
  const int mo = (lane >> 4) << 3;
#pragma unroll
  for (int i = 0; i < 2; ++i)
#pragma unroll
    for (int jt = 0; jt < 4; ++jt)
#pragma unroll
      for (int j = 0; j < 8; ++j)
        out[(size_t)(rg * 32 + i * 16 + mo + j) * D_MODEL + wn0 + jt * 16 + n] =
            acc[i][jt][j];
}

// ---------------------------------------------------------------------------
extern "C" void kernel_launch(void* const* d_in, const int* in_sizes, int n_in,
                              void* d_out, int out_size, void* d_ws, size_t ws_size,
                              hipStream_t stream) {
  (void)in_sizes; (void)n_in; (void)out_size; (void)ws_size;
  const float* x  = (const float*)d_in[0];
  const float* Wq = (const float*)d_in[1];
  const float* Wk = (const float*)d_in[2];
  const float* Wv = (const float*)d_in[3];
  const float* Wo = (const float*)d_in[4];
  const float* dl = (const float*)d_in[5];
  const float* nw = (const float*)d_in[6];
  float* out = (float*)d_out;

  float* ws  = (float*)d_ws;
  float* Q   = ws;                               // 4096*64
  float* K   = Q   + (size_t)ROWS * MEM_DIM;     // 4096*64
  float* V   = K   + (size_t)ROWS * MEM_DIM;     // 4096*512
  float* Num = V   + (size_t)ROWS * D_MODEL;     // 4096*512
  float* Den = Num + (size_t)ROWS * D_MODEL;     // 4096
  float* Un  = Den + (size_t)ROWS;               // 4096*512

  syn_qkv_kernel<<<dim3(10, 32), 128, 0, stream>>>(x, Wq, Wk, Wv, Q, K, V);
  syn_scan_kernel<<<dim3(D_MODEL / DBLK, BATCH), 128, 0, stream>>>(Q, K, V, dl, Num, Den);
  syn_norm_kernel<<<dim3(ROWS), 128, 0, stream>>>(Num, Den, nw, Un);
  syn_oproj_kernel<<<dim3(8, 32), 128, 0, stream>>>(Un, Wo, out);
}